// EmissionModel_37117107372253
// MI455X (gfx1250) — compile-verified
//
#include <hip/hip_runtime.h>
#include <math.h>

// Problem constants (from reference): B=8192, N=64, D=128
#define DDIM   128
#define NCOMP  64
#define BATCH  8192
#define BT     128          // batch tile per workgroup
#define LDA    132          // padded LDS row stride for Linv (132 % 64 = 4 banks -> conflict-free)

typedef __attribute__((ext_vector_type(2))) float v2f;
typedef __attribute__((ext_vector_type(8))) float v8f;

// Match the builtin's expected pointee type exactly (per hipcc diagnostic):
// '__attribute__((__vector_size__(4 * sizeof(int)))) int *'
typedef int v4i __attribute__((vector_size(16)));
typedef __attribute__((address_space(1))) v4i* g_v4i_ptr;   // global
typedef __attribute__((address_space(3))) v4i* l_v4i_ptr;   // LDS

#if defined(__gfx1250__) && __has_builtin(__builtin_amdgcn_global_load_async_to_lds_b128)
#define HAVE_ASYNC_LDS 1
#else
#define HAVE_ASYNC_LDS 0
#endif

__device__ __forceinline__ void wait_asynccnt0() {
#if __has_builtin(__builtin_amdgcn_s_wait_asynccnt)
  __builtin_amdgcn_s_wait_asynccnt(0);
#else
  asm volatile("s_wait_asynccnt 0x0" ::: "memory");
#endif
}

// ---------------------------------------------------------------------------
// Kernel A: per-component lower-triangular inverse (exp-reparameterized diag)
// and log-determinant. One block per component, thread j owns column j.
// LDS: Lsh[D*D] | Ish[D*D] | diag[D] | invd[D]  (132,096 B dynamic)
// ---------------------------------------------------------------------------
__global__ __launch_bounds__(DDIM) void trinv_logdet_kernel(
    const float* __restrict__ chol, float* __restrict__ linv,
    float* __restrict__ logdet) {
  extern __shared__ float smem[];
  float* Lsh  = smem;                    // [D][D]
  float* Ish  = smem + DDIM * DDIM;      // [D][D]
  float* diag = smem + 2 * DDIM * DDIM;  // [D]
  float* invd = diag + DDIM;             // [D]

  const int n = blockIdx.x;
  const int j = threadIdx.x;             // column owned by this thread
  const float* Ln = chol + (size_t)n * DDIM * DDIM;

  // Load column j of L: tril + exp on diagonal (coalesced across threads per row)
  for (int i = 0; i < DDIM; ++i) {
    float v = Ln[i * DDIM + j];
    float o;
    if (i < j) {
      o = 0.0f;
    } else if (i == j) {
      diag[j] = v;          // raw diagonal for logdet
      o = expf(v);
    } else {
      o = v;
    }
    Lsh[i * DDIM + j] = o;
  }
  __syncthreads();

  invd[j] = 1.0f / Lsh[j * DDIM + j];
  if (j == 0) {
    float s = 0.0f;
    for (int i = 0; i < DDIM; ++i) s += diag[i];
    logdet[n] = 2.0f * s;   // log det Sigma = 2 * sum(raw diag)
  }
  __syncthreads();

  // Forward substitution: L * y = e_j  ->  y = column j of L^{-1}.
  // All Ish reads below are this thread's own prior writes.
  for (int i = j; i < DDIM; ++i) {
    float s = 0.0f;
    for (int k = j; k < i; ++k) s += Lsh[i * DDIM + k] * Ish[k * DDIM + j];
    Ish[i * DDIM + j] = ((i == j ? 1.0f : 0.0f) - s) * invd[i];
  }

  // Write row-major Linv[n][i][j]; explicit zeros above the diagonal so the
  // GEMM kernel can consume dense tiles.
  float* dst = linv + (size_t)n * DDIM * DDIM;
  for (int i = 0; i < DDIM; ++i)
    dst[i * DDIM + j] = (i < j) ? 0.0f : Ish[i * DDIM + j];
}

// ---------------------------------------------------------------------------
// Kernel B: fused  quad[b,n] = || Linv_n (x_b - mu_n) ||^2  via f32 WMMA,
// then log_p = -0.5*(log_c + logdet[n] + quad).
//
// Block = (component n, batch tile of 128). 256 threads = 8 wave32s.
// Each wave owns a 16-wide batch strip and ALL 8 M-tiles (full D=128 rows),
// so the sum-of-squares over M reduces inside the wave (shfl_xor 16).
//
// Linv is staged straight into (padded) LDS with GLOBAL_LOAD_ASYNC_TO_LDS_B128
// (ASYNCcnt path) when available; the delta tile stays on the manual path
// because it fuses the x - mu subtraction.
// ---------------------------------------------------------------------------
__global__ __launch_bounds__(256) void emission_logp_kernel(
    const float* __restrict__ x, const float* __restrict__ means,
    const float* __restrict__ linv, const float* __restrict__ logdet,
    float* __restrict__ out) {
  extern __shared__ float smem[];
  float* As = smem;               // Linv tile [D][LDA] (padded)
  float* Bs = smem + DDIM * LDA;  // Delta tile [D(K)][BT]

  const int n   = blockIdx.x;     // component
  const int bt  = blockIdx.y;     // batch tile index
  const int tid = threadIdx.x;

  // ---- Stage Linv_n into LDS (row-major, padded stride) -------------------
#if HAVE_ASYNC_LDS
  {
    const float* src = linv + (size_t)n * DDIM * DDIM;
    #pragma unroll
    for (int it = 0; it < 16; ++it) {          // 4096 float4 / 256 threads
      int idx  = it * 256 + tid;               // float4 index
      int row  = idx >> 5;                     // 32 float4 per 128-float row
      int col4 = idx & 31;
      const float* g = src + (size_t)idx * 4;          // 16B aligned
      float*       l = As + row * LDA + col4 * 4;      // 16B aligned
      __builtin_amdgcn_global_load_async_to_lds_b128(
          (g_v4i_ptr)g, (l_v4i_ptr)l, 0, 0);
    }
  }
#else
  {
    const float4* src = (const float4*)(linv + (size_t)n * DDIM * DDIM);
    #pragma unroll
    for (int it = 0; it < 16; ++it) {
      int idx  = it * 256 + tid;
      int row  = idx >> 5;
      int col4 = idx & 31;
      float4 v = src[idx];
      float* d = As + row * LDA + col4 * 4;
      d[0] = v.x; d[1] = v.y; d[2] = v.z; d[3] = v.w;
    }
  }
#endif
  // ---- Stage Delta = (x - mu)^T into LDS: Bs[d][b_local] ------------------
  {
    const float4* xb = (const float4*)(x + (size_t)bt * BT * DDIM);
    const float4* mu = (const float4*)(means + (size_t)n * DDIM);
    #pragma unroll
    for (int it = 0; it < 16; ++it) {          // BT*D/4 = 4096 float4
      int idx = it * 256 + tid;
      int b   = idx >> 5;                      // 32 float4 per x row
      int d4  = idx & 31;
      float4 v = xb[idx];
      float4 m = mu[d4];
      Bs[(d4 * 4 + 0) * BT + b] = v.x - m.x;   // consecutive b -> consecutive
      Bs[(d4 * 4 + 1) * BT + b] = v.y - m.y;   // banks, but a wave's lanes share b (4*BT stride: bank conflicts)
      Bs[(d4 * 4 + 2) * BT + b] = v.z - m.z;
      Bs[(d4 * 4 + 3) * BT + b] = v.w - m.w;
    }
  }
#if HAVE_ASYNC_LDS
  wait_asynccnt0();               // async tile landed in LDS for this wave
#endif
  __syncthreads();

  // ---- WMMA main loop (fully unrolled, constant indices everywhere) -------
  const int wave    = tid >> 5;
  const int lane    = tid & 31;
  const int nl      = lane & 15;     // N (batch col) for B/C/D, M lane for A
  const int khalf   = lane >> 4;     // lane-half selects K pair {0,1} vs {2,3}
  const int colbase = wave * 16;

  const float* Abase = As + nl * LDA + 2 * khalf;             // + mt*16*LDA + kb
  const float* Bbase = Bs + (2 * khalf) * BT + colbase + nl;  // + kb*BT

  v8f acc[8];
  {
    v8f z = {};
    #pragma unroll
    for (int mt = 0; mt < 8; ++mt) acc[mt] = z;
  }

  #pragma unroll
  for (int ks = 0; ks < 32; ++ks) {            // K = 128 in steps of 4
    const int kb = ks * 4;
    v2f bfrag;
    bfrag.x = Bbase[(kb + 0) * BT];
    bfrag.y = Bbase[(kb + 1) * BT];
    #pragma unroll
    for (int mt = 0; mt < 8; ++mt) {
      if (mt >= (ks >> 2)) {                   // compile-time triangular skip
        v2f afrag = *(const v2f*)(Abase + mt * 16 * LDA + kb);
        acc[mt] = __builtin_amdgcn_wmma_f32_16x16x4_f32(
            /*neg_a=*/false, afrag, /*neg_b=*/false, bfrag,
            /*c_mod=*/(short)0, acc[mt], /*reuse_a=*/false, /*reuse_b=*/false);
      }
    }
  }

  // ---- Epilogue: quad = sum over M of y^2, pair-reduce lane halves --------
  float q = 0.0f;
  #pragma unroll
  for (int mt = 0; mt < 8; ++mt) {
    v8f a = acc[mt];
    #pragma unroll
    for (int r = 0; r < 8; ++r) q += a[r] * a[r];
  }
  q += __shfl_xor(q, 16, 32);   // lane l (+) lane l+16: full 128-row sum

  if (lane < 16) {
    const float log_c = 235.2482645f;          // 128 * ln(2*pi)
    const float ld = logdet[n];
    const int b = bt * BT + colbase + nl;
    out[(size_t)b * NCOMP + n] = -0.5f * (log_c + ld + q);
  }
}

// ---------------------------------------------------------------------------
extern "C" void kernel_launch(void* const* d_in, const int* in_sizes, int n_in,
                              void* d_out, int out_size, void* d_ws, size_t ws_size,
                              hipStream_t stream) {
  const float* x     = (const float*)d_in[0];   // (8192, 128)
  const float* means = (const float*)d_in[1];   // (64, 128)
  const float* chol  = (const float*)d_in[2];   // (64, 128, 128)
  float* out = (float*)d_out;                   // (8192, 64)

  float* logdet = (float*)d_ws;                 // 64 floats (padded to 256)
  float* linvW  = logdet + 256;                 // 64*128*128 floats = 4 MB

  const size_t smemA = (size_t)(2 * DDIM * DDIM + 2 * DDIM) * sizeof(float); // 132,096 B
  const size_t smemB = (size_t)(DDIM * LDA + DDIM * BT) * sizeof(float);     // 133,120 B

  // Opt in to >64KB dynamic LDS (320 KB per WGP on CDNA5).
  (void)hipFuncSetAttribute((const void*)trinv_logdet_kernel,
                            hipFuncAttributeMaxDynamicSharedMemorySize, (int)smemA);
  (void)hipFuncSetAttribute((const void*)emission_logp_kernel,
                            hipFuncAttributeMaxDynamicSharedMemorySize, (int)smemB);

  trinv_logdet_kernel<<<dim3(NCOMP), dim3(DDIM), smemA, stream>>>(
      chol, linvW, logdet);

  emission_logp_kernel<<<dim3(NCOMP, BATCH / BT), dim3(256), smemB, stream>>>(
      x, means, linvW, logdet, out);
}